// GENConv_936302871063
// MI455X (gfx1250) — compile-verified
//
#include <hip/hip_runtime.h>
#include <hip/hip_bf16.h>

#define N_NODES 100000
#define N_EDGES 1600000
#define DIM     64
#define BETA    1.0f

typedef __attribute__((ext_vector_type(2))) float v2f;
typedef __attribute__((ext_vector_type(8))) float v8f;

// IEEE-monotone float atomic max: for val>=0 signed-int max matches float order;
// for val<0 unsigned min matches float order. Init must be -inf (0xFF800000).
__device__ __forceinline__ void atomic_max_f32(float* addr, float val) {
    if (val >= 0.0f) atomicMax((int*)addr, __float_as_int(val));
    else             atomicMin((unsigned int*)addr, __float_as_uint(val));
}

// u[i] = mean_j Wm[i][j]  (so that mean(x@Wm,axis=-1) = x . u), u[DIM] = mean(bm)
__global__ void k_prep(const float* __restrict__ Wm, const float* __restrict__ bm,
                       float* __restrict__ u) {
    int i = threadIdx.x;
    if (i < DIM) {
        float acc = 0.f;
        for (int j = 0; j < DIM; ++j) acc += Wm[i * DIM + j];
        u[i] = acc * (1.0f / DIM);
    }
    if (i == 0) {
        float acc = 0.f;
        for (int j = 0; j < DIM; ++j) acc += bm[j];
        u[DIM] = acc * (1.0f / DIM);
    }
}

// Per-node score s[n] = BETA*(x_n . u + c); init smax=-inf, ssum=0, zero h (=d_out)
__global__ void k_node(const float* __restrict__ x, const float* __restrict__ u,
                       float* __restrict__ s, float* __restrict__ smax,
                       float* __restrict__ ssum, float* __restrict__ out) {
    int n = blockIdx.x * blockDim.x + threadIdx.x;
    if (n >= N_NODES) return;
    const float c = u[DIM];
    const float4* x4 = (const float4*)(x + (size_t)n * DIM);
    const float4* u4 = (const float4*)u;
    float acc = 0.f;
#pragma unroll
    for (int j = 0; j < DIM / 4; ++j) {
        float4 a = x4[j], b = u4[j];
        acc += a.x * b.x + a.y * b.y + a.z * b.z + a.w * b.w;
    }
    s[n]    = BETA * (acc + c);
    smax[n] = -__builtin_inff();
    ssum[n] = 0.f;
    float4* o4 = (float4*)(out + (size_t)n * DIM);
    float4 z; z.x = z.y = z.z = z.w = 0.f;
#pragma unroll
    for (int j = 0; j < DIM / 4; ++j) o4[j] = z;
}

__global__ void k_edge_max(const int* __restrict__ row, const int* __restrict__ col,
                           const float* __restrict__ s, float* __restrict__ smax) {
    int e = blockIdx.x * blockDim.x + threadIdx.x;
    if (e >= N_EDGES) return;
    atomic_max_f32(&smax[col[e]], s[row[e]]);
}

__global__ void k_edge_sum(const int* __restrict__ row, const int* __restrict__ col,
                           const float* __restrict__ s, const float* __restrict__ smax,
                           float* __restrict__ ssum) {
    int e = blockIdx.x * blockDim.x + threadIdx.x;
    if (e >= N_EDGES) return;
    int cidx = col[e];
    atomicAdd(&ssum[cidx], __expf(s[row[e]] - smax[cidx]));
}

// h[col] += w_e * x[row];  16 lanes per edge, one float4 per lane (x is L2-resident).
__global__ void k_scatter(const int* __restrict__ row, const int* __restrict__ col,
                          const float* __restrict__ s, const float* __restrict__ smax,
                          const float* __restrict__ ssum, const float* __restrict__ x,
                          float* __restrict__ out) {
    long long gid = (long long)blockIdx.x * blockDim.x + threadIdx.x;
    int e = (int)(gid >> 4);
    int t = (int)(gid & 15);
    if (e >= N_EDGES) return;
    int r = row[e], cidx = col[e];
    float w = __expf(s[r] - smax[cidx]) / ssum[cidx];
    float4 xa = ((const float4*)(x + (size_t)r * DIM))[t];
    float* o = out + (size_t)cidx * DIM + t * 4;
    atomicAdd(o + 0, w * xa.x);
    atomicAdd(o + 1, w * xa.y);
    atomicAdd(o + 2, w * xa.z);
    atomicAdd(o + 3, w * xa.w);
}

// out = h@Wm + 1{has edges}*bm + x@Wr + br  via V_WMMA_F32_16X16X4_F32.
// Block = 128 threads = 4 waves; block owns 16 node rows; wave w owns cols [16w,16w+16).
// h tile staged through LDS so the in-place overwrite of d_out is hazard-free.
__global__ void k_out(const float* __restrict__ x, const float* __restrict__ Wm,
                      const float* __restrict__ bm, const float* __restrict__ Wr,
                      const float* __restrict__ br, const float* __restrict__ ssum,
                      float* __restrict__ out) {
    __shared__ float hA[16 * DIM];
    const int rm  = blockIdx.x * 16;
    const int tid = threadIdx.x;

    // cooperative load: 16x64 = 256 float4, 2 per thread
    {
        const float4* src = (const float4*)(out + (size_t)rm * DIM);
        float4* dst = (float4*)hA;
        dst[tid]       = src[tid];
        dst[tid + 128] = src[tid + 128];
    }
    __syncthreads();

    const int wave = tid >> 5;
    const int lane = tid & 31;
    const int hf   = lane >> 4;   // 0: K pair {0,1}, 1: K pair {2,3}
    const int lh   = lane & 15;   // M (for A) / N (for B,C,D)
    const int cn   = wave * 16;

    v8f acc = {};
#pragma unroll
    for (int k0 = 0; k0 < DIM; k0 += 4) {
        const int kk = k0 + 2 * hf;
        // h @ Wm
        v2f a, b;
        a.x = hA[lh * DIM + kk];
        a.y = hA[lh * DIM + kk + 1];
        b.x = Wm[(size_t)kk * DIM + cn + lh];
        b.y = Wm[(size_t)(kk + 1) * DIM + cn + lh];
        acc = __builtin_amdgcn_wmma_f32_16x16x4_f32(false, a, false, b,
                                                    (short)0, acc, false, false);
        // x @ Wr
        v2f a2, b2;
        a2.x = x[(size_t)(rm + lh) * DIM + kk];
        a2.y = x[(size_t)(rm + lh) * DIM + kk + 1];
        b2.x = Wr[(size_t)kk * DIM + cn + lh];
        b2.y = Wr[(size_t)(kk + 1) * DIM + cn + lh];
        acc = __builtin_amdgcn_wmma_f32_16x16x4_f32(false, a2, false, b2,
                                                    (short)0, acc, false, false);
    }

    const float bmv = bm[cn + lh];
    const float brv = br[cn + lh];
#pragma unroll
    for (int v = 0; v < 8; ++v) {
        int rowm = v + 8 * hf;                       // C/D layout: lanes 16-31 hold M+8
        float flag = (ssum[rm + rowm] > 0.f) ? 1.f : 0.f;
        out[(size_t)(rm + rowm) * DIM + cn + lh] = acc[v] + flag * bmv + brv;
    }
}

extern "C" void kernel_launch(void* const* d_in, const int* in_sizes, int n_in,
                              void* d_out, int out_size, void* d_ws, size_t ws_size,
                              hipStream_t stream) {
    const float* x  = (const float*)d_in[0];
    const int*   ei = (const int*)d_in[1];
    const float* Wm = (const float*)d_in[2];
    const float* bm = (const float*)d_in[3];
    const float* Wr = (const float*)d_in[4];
    const float* br = (const float*)d_in[5];
    float* out = (float*)d_out;

    const int* row = ei;
    const int* col = ei + N_EDGES;

    float* s    = (float*)d_ws;          // N
    float* smax = s + N_NODES;           // N
    float* ssum = smax + N_NODES;        // N
    float* u    = ssum + N_NODES;        // DIM+1

    k_prep<<<1, 64, 0, stream>>>(Wm, bm, u);
    k_node<<<(N_NODES + 255) / 256, 256, 0, stream>>>(x, u, s, smax, ssum, out);
    k_edge_max<<<(N_EDGES + 255) / 256, 256, 0, stream>>>(row, col, s, smax);
    k_edge_sum<<<(N_EDGES + 255) / 256, 256, 0, stream>>>(row, col, s, smax, ssum);
    long long tot = (long long)N_EDGES * 16;
    k_scatter<<<(int)((tot + 255) / 256), 256, 0, stream>>>(row, col, s, smax, ssum, x, out);
    k_out<<<N_NODES / 16, 128, 0, stream>>>(x, Wm, bm, Wr, br, ssum, out);
}